// LightEncoder_80693845557943
// MI455X (gfx1250) — compile-verified
//
#include <hip/hip_runtime.h>
#include <hip/hip_bf16.h>

typedef __attribute__((ext_vector_type(2))) float v2f;
typedef __attribute__((ext_vector_type(8))) float v8f;

#define FDIM 64

// ---------------------------------------------------------------- zero scratch
__global__ void zero_kernel(float* __restrict__ p, long n) {
    long i = (long)blockIdx.x * blockDim.x + threadIdx.x;
    if (i < n) p[i] = 0.0f;
}

// ---------------------------------------------------------------- degrees
__global__ void degree_kernel(const int* __restrict__ src, const int* __restrict__ dst,
                              float* __restrict__ deg_out, float* __restrict__ deg_in, int E) {
    int e = blockIdx.x * blockDim.x + threadIdx.x;
    if (e < E) {
        atomicAdd(&deg_out[src[e]], 1.0f);
        atomicAdd(&deg_in[dst[e]], 1.0f);
    }
}

// deg -> rsqrt(clamp(deg,1)) in place, over both deg arrays (2N contiguous floats)
__global__ void norm_kernel(float* __restrict__ deg, int n2) {
    int i = blockIdx.x * blockDim.x + threadIdx.x;
    if (i < n2) deg[i] = rsqrtf(fmaxf(deg[i], 1.0f));
}

// ---------------------------------------------------------------- edge scatter
// 16 lanes per edge, 4 features per lane (float4 gather, 4 f32 atomics).
__global__ void scatter_kernel(const int* __restrict__ src, const int* __restrict__ dst,
                               const float* __restrict__ x, const float* __restrict__ norm_src,
                               float* __restrict__ agg, int E) {
    int gid = blockIdx.x * blockDim.x + threadIdx.x;
    int e  = gid >> 4;
    int f4 = (gid & 15) * 4;
    if (e < E) {
        int s = src[e];
        int d = dst[e];
        float ns = norm_src[s];
        const float4 v = *(const float4*)(x + (long)s * FDIM + f4);
        float* ap = agg + (long)d * FDIM + f4;
        atomicAdd(ap + 0, v.x * ns);
        atomicAdd(ap + 1, v.y * ns);
        atomicAdd(ap + 2, v.z * ns);
        atomicAdd(ap + 3, v.w * ns);
    }
}

// ---------------------------------------------------------------- fused WMMA GEMM
// out = (agg * norm_dst) @ W + x @ Wr + (b + br)
// One wave32 per 16-row strip; 4 column tiles (N=0,16,32,48); K in steps of 4
// via v_wmma_f32_16x16x4_f32.
//
// A 16x4 f32 layout:  lanes 0-15 = rows M0..15; VGPR0 = {K0 | K2}, VGPR1 = {K1 | K3}
// B 4x16  f32 layout:  VGPR0 = row {K0 | K2} across N, VGPR1 = row {K1 | K3}
// C/D 16x16 f32:       VGPR r -> row (r + 8*half), col = lane&15
__global__ void __launch_bounds__(256)
gconv_gemm_kernel(const float* __restrict__ agg, const float* __restrict__ x,
                  const float* __restrict__ W, const float* __restrict__ Wr,
                  const float* __restrict__ b, const float* __restrict__ br,
                  const float* __restrict__ norm_dst, float* __restrict__ out, int n) {
    const int lane = threadIdx.x & 31;
    const int wave = threadIdx.x >> 5;
    const int wid  = blockIdx.x * (blockDim.x >> 5) + wave;
    const int row_base = wid * 16;
    if (row_base >= n) return;            // wave-uniform: EXEC stays all-ones

    const int half = lane >> 4;           // 0: lanes 0-15, 1: lanes 16-31
    const int l16  = lane & 15;
    const int arow = row_base + l16;      // A-operand row for this lane
    const float nd = norm_dst[arow];

    v8f acc[4] = {v8f{}, v8f{}, v8f{}, v8f{}};

    // ---- (agg * norm_dst) @ W
    for (int k0 = 0; k0 < FDIM; k0 += 4) {
        const int ka = k0 + half * 2;
        v2f a;
        a.x = agg[(long)arow * FDIM + ka + 0] * nd;
        a.y = agg[(long)arow * FDIM + ka + 1] * nd;
#pragma unroll
        for (int t = 0; t < 4; ++t) {
            const int ncol = t * 16 + l16;
            v2f bm;
            bm.x = W[(ka + 0) * FDIM + ncol];
            bm.y = W[(ka + 1) * FDIM + ncol];
            acc[t] = __builtin_amdgcn_wmma_f32_16x16x4_f32(
                false, a, false, bm, (short)0, acc[t], false, false);
        }
    }

    // ---- residual: x @ Wr
    for (int k0 = 0; k0 < FDIM; k0 += 4) {
        const int ka = k0 + half * 2;
        v2f a;
        a.x = x[(long)arow * FDIM + ka + 0];
        a.y = x[(long)arow * FDIM + ka + 1];
#pragma unroll
        for (int t = 0; t < 4; ++t) {
            const int ncol = t * 16 + l16;
            v2f bm;
            bm.x = Wr[(ka + 0) * FDIM + ncol];
            bm.y = Wr[(ka + 1) * FDIM + ncol];
            acc[t] = __builtin_amdgcn_wmma_f32_16x16x4_f32(
                false, a, false, bm, (short)0, acc[t], false, false);
        }
    }

    // ---- bias + store (C/D layout)
#pragma unroll
    for (int t = 0; t < 4; ++t) {
        const int col = t * 16 + l16;
        const float bias = b[col] + br[col];
#pragma unroll
        for (int r = 0; r < 8; ++r) {
            const int row = row_base + r + half * 8;
            out[(long)row * FDIM + col] = acc[t][r] + bias;
        }
    }
}

// ---------------------------------------------------------------- launcher
extern "C" void kernel_launch(void* const* d_in, const int* in_sizes, int n_in,
                              void* d_out, int out_size, void* d_ws, size_t ws_size,
                              hipStream_t stream) {
    const float* x  = (const float*)d_in[0];
    const int*   src = (const int*)d_in[1];
    const int*   dst = (const int*)d_in[2];
    const float* W  = (const float*)d_in[3];
    const float* b  = (const float*)d_in[4];
    const float* Wr = (const float*)d_in[5];
    const float* br = (const float*)d_in[6];
    float* out = (float*)d_out;

    const int N = in_sizes[0] / FDIM;   // 50000
    const int E = in_sizes[1];          // 800000

    // workspace layout (floats): [deg_out N][deg_in N][agg N*64]
    float* ws       = (float*)d_ws;
    float* deg_out  = ws;               // -> norm_src after norm_kernel
    float* deg_in   = ws + N;           // -> norm_dst after norm_kernel
    float* agg      = ws + 2 * (long)N;

    const int TB = 256;

    // 1) zero deg + agg
    long nz = 2L * N + (long)N * FDIM;
    zero_kernel<<<(int)((nz + TB - 1) / TB), TB, 0, stream>>>(ws, nz);

    // 2) degrees
    degree_kernel<<<(E + TB - 1) / TB, TB, 0, stream>>>(src, dst, deg_out, deg_in, E);

    // 3) norms in place
    norm_kernel<<<(2 * N + TB - 1) / TB, TB, 0, stream>>>(ws, 2 * N);

    // 4) edge scatter: 16 lanes per edge
    long sthreads = (long)E * 16;
    scatter_kernel<<<(int)((sthreads + TB - 1) / TB), TB, 0, stream>>>(
        src, dst, x, deg_out, agg, E);

    // 5) fused WMMA GEMM + bias + residual: one wave per 16 rows, 8 waves/block
    int row_tiles = (N + 15) / 16;                 // 3125
    int blocks = (row_tiles + 7) / 8;              // 8 waves per 256-thread block
    gconv_gemm_kernel<<<blocks, TB, 0, stream>>>(agg, x, W, Wr, b, br, deg_in, out, N);
}